// Loss_21096879358429
// MI455X (gfx1250) — compile-verified
//
#include <hip/hip_runtime.h>
#include <math.h>

// Problem constants (match reference)
#define BSZ   4096
#define TLEN  512
#define KST   7
#define NCHUNK 8
#define CHLEN  64   // timesteps per chunk (t in [1 + c*64, 64 + c*64])

typedef float v2f __attribute__((ext_vector_type(2)));
typedef float v8f __attribute__((ext_vector_type(8)));

// ds_swizzle group-of-32 patterns: offset = {xor[14:10], or[9:5], and[4:0]}
template <int PAT>
__device__ __forceinline__ float swz(float x) {
    return __int_as_float(__builtin_amdgcn_ds_swizzle(__float_as_int(x), PAT));
}
#define SWAPX16 0x401F  // xor 0x10, and 0x1f
#define XOR1    0x041F
#define XOR2    0x081F
#define XOR4    0x101F

// ---------------------------------------------------------------------------
// Phase 1: per (4-batch group, chunk) wave computes the chunk's 7x7 transfer
// matrix in scaled-exp domain using v_wmma_f32_16x16x4_f32, plus the path
// score partial for the chunk's time range.
//
// Lane map: lane L -> batch index bi = L>>3 (0..3), matrix row a = L&7 (7=pad).
// After each WMMA chain, lane L's acc regs 0..6 hold states 0..6 of row a of
// batch bi -- all post-processing is lane-local.
// ---------------------------------------------------------------------------
__global__ void __launch_bounds__(256) crf_phase1(
    const float* __restrict__ emissions,   // [B,T,K]
    const int*   __restrict__ mask,        // [B,T]
    const int*   __restrict__ tags,        // [B,T]
    const float* __restrict__ transitions, // [K,K]
    const float* __restrict__ end_tr,      // [K]
    float* __restrict__ wsG,               // [NCHUNK,B,8,8]  (cols 0..6 = G, col 7 = row log-offset)
    float* __restrict__ wsScore)           // [NCHUNK,B]
{
    const int lane  = threadIdx.x & 31;
    const int wave  = blockIdx.x * (blockDim.x >> 5) + (threadIdx.x >> 5);
    const int bg    = wave % (BSZ / 4);
    const int chunk = wave / (BSZ / 4);
    const int bi    = lane >> 3;        // which of the 4 batches
    const int a     = lane & 7;         // matrix row (7 = padding row)
    const int batch = bg * 4 + bi;
    const bool hi   = lane >= 16;

    // ---- Constant A operand: expTrans^T placed block-diagonally in 16x16.
    // A[j][k]: reg R=k&1, lane = j + 16*(k>>1) per 16x4 chunk c (k = 4c+R+2*(lane>>4)).
    const int aj = lane & 15;
    float Av[4][2];
#pragma unroll
    for (int c = 0; c < 4; ++c) {
#pragma unroll
        for (int R = 0; R < 2; ++R) {
            const int ak = 4 * c + R + ((lane >> 4) << 1);
            float v = 0.0f;
            if (aj < 7 && ak < 7)                               v = __expf(transitions[ak * KST + aj]);
            else if (aj >= 8 && aj < 15 && ak >= 8 && ak < 15)  v = __expf(transitions[(ak - 8) * KST + (aj - 8)]);
            Av[c][R] = v;
        }
    }
    const v2f A0 = {Av[0][0], Av[0][1]};
    const v2f A1 = {Av[1][0], Av[1][1]};
    const v2f A2 = {Av[2][0], Av[2][1]};
    const v2f A3 = {Av[3][0], Av[3][1]};

    // ---- Evolving chunk matrix, scaled-exp domain. G starts as identity.
    float g0 = (a == 0) ? 1.f : 0.f, g1 = (a == 1) ? 1.f : 0.f, g2 = (a == 2) ? 1.f : 0.f;
    float g3 = (a == 3) ? 1.f : 0.f, g4 = (a == 4) ? 1.f : 0.f, g5 = (a == 5) ? 1.f : 0.f;
    float g6 = (a == 6) ? 1.f : 0.f;
    float mcar = 0.0f;                  // per-row log offset

    const int tbase = 1 + chunk * CHLEN;
    const size_t ebase = (size_t)batch * TLEN * KST;

#pragma unroll 2
    for (int s = 0; s < CHLEN; ++s) {
        const int t = tbase + s;
        if (t >= TLEN) break;           // uniform across wave

        const size_t eo = ebase + (size_t)t * KST;
        const float e0 = __expf(emissions[eo + 0]);
        const float e1 = __expf(emissions[eo + 1]);
        const float e2 = __expf(emissions[eo + 2]);
        const float e3 = __expf(emissions[eo + 3]);
        const float e4 = __expf(emissions[eo + 4]);
        const float e5 = __expf(emissions[eo + 5]);
        const float e6 = __expf(emissions[eo + 6]);
        const int   mt = mask[batch * TLEN + t];

        // Build B operands (k rows 0..6 = batches 0/1, rows 8..14 = batches 2/3).
        const float s0 = swz<SWAPX16>(g0), s1 = swz<SWAPX16>(g1), s2 = swz<SWAPX16>(g2);
        const float s3 = swz<SWAPX16>(g3), s4 = swz<SWAPX16>(g4), s5 = swz<SWAPX16>(g5);
        const float s6 = swz<SWAPX16>(g6);
        const v2f B0 = {hi ? s2 : g0, hi ? s3 : g1};
        const v2f B1 = {hi ? s6 : g4, hi ? 0.f : g5};
        const v2f B2 = {hi ? g2 : s0, hi ? g3 : s1};
        const v2f B3 = {hi ? g6 : s4, hi ? 0.f : s5};

        v8f acc = {};
        acc = __builtin_amdgcn_wmma_f32_16x16x4_f32(false, A0, false, B0, (short)0, acc, false, false);
        acc = __builtin_amdgcn_wmma_f32_16x16x4_f32(false, A1, false, B1, (short)0, acc, false, false);
        acc = __builtin_amdgcn_wmma_f32_16x16x4_f32(false, A2, false, B2, (short)0, acc, false, false);
        acc = __builtin_amdgcn_wmma_f32_16x16x4_f32(false, A3, false, B3, (short)0, acc, false, false);

        // Lane-local: apply emission, renormalize row, blend with mask.
        const float u0 = acc[0] * e0, u1 = acc[1] * e1, u2 = acc[2] * e2, u3 = acc[3] * e3;
        const float u4 = acc[4] * e4, u5 = acc[5] * e5, u6 = acc[6] * e6;
        float M = fmaxf(fmaxf(fmaxf(u0, u1), fmaxf(u2, u3)), fmaxf(fmaxf(u4, u5), u6));
        M = fmaxf(M, 1e-38f);
        const float inv = __builtin_amdgcn_rcpf(M);
        const float lm  = __logf(M);
        const bool keep = (mt != 0);
        g0 = keep ? u0 * inv : g0;  g1 = keep ? u1 * inv : g1;
        g2 = keep ? u2 * inv : g2;  g3 = keep ? u3 * inv : g3;
        g4 = keep ? u4 * inv : g4;  g5 = keep ? u5 * inv : g5;
        g6 = keep ? u6 * inv : g6;
        mcar += keep ? lm : 0.0f;
    }

    // ---- Path-score partial: 8 lanes of a batch split the 64 steps.
    float sc = 0.0f;
#pragma unroll
    for (int i = 0; i < 8; ++i) {
        const int t = tbase + a + 8 * i;
        if (t < TLEN) {
            const int mt = mask[batch * TLEN + t];
            if (mt) {
                const int cur  = tags[batch * TLEN + t];
                const int prev = tags[batch * TLEN + t - 1];
                sc += transitions[prev * KST + cur] + emissions[ebase + (size_t)t * KST + cur];
                const int mnext = (t + 1 < TLEN) ? mask[batch * TLEN + t + 1] : 0;
                if (!mnext) sc += end_tr[cur];   // last valid step -> end transition
            }
        }
    }
    sc += swz<XOR1>(sc);
    sc += swz<XOR2>(sc);
    sc += swz<XOR4>(sc);
    if (a == 0) wsScore[chunk * BSZ + batch] = sc;

    // ---- Store chunk matrix (row-per-lane, fully coalesced 32B/lane).
    float* dst = wsG + (((size_t)chunk * BSZ + batch) * 8 + a) * 8;
    *(float4*)(dst + 0) = make_float4(g0, g1, g2, g3);
    *(float4*)(dst + 4) = make_float4(g4, g5, g6, mcar);
}

// ---------------------------------------------------------------------------
// Phase 2: per batch, fold alpha0 through the 8 chunk matrices (logsumexp in
// log domain, clamped), add end transitions, assemble score, write log_z-score.
// ---------------------------------------------------------------------------
__global__ void __launch_bounds__(256) crf_phase2(
    const float* __restrict__ emissions,
    const int*   __restrict__ mask,
    const int*   __restrict__ tags,
    const float* __restrict__ start_tr,
    const float* __restrict__ end_tr,
    const float* __restrict__ wsG,
    const float* __restrict__ wsScore,
    float* __restrict__ perBatch)
{
    const int b = blockIdx.x * blockDim.x + threadIdx.x;
    if (b >= BSZ) return;

    float alpha[KST];
#pragma unroll
    for (int j = 0; j < KST; ++j)
        alpha[j] = start_tr[j] + emissions[(size_t)b * TLEN * KST + j];

    for (int c = 0; c < NCHUNK; ++c) {
        const float* g = wsG + ((size_t)c * BSZ + b) * 64;
        float na[KST];
#pragma unroll
        for (int j = 0; j < KST; ++j) {
            float tv[KST];
            float mx = -3.0e38f;
#pragma unroll
            for (int r = 0; r < KST; ++r) {
                const float ga = g[r * 8 + j];
                const float lg = (ga > 0.0f) ? __logf(ga) : -1e30f;
                float t2 = alpha[r] + g[r * 8 + 7] + lg;
                t2 = fmaxf(t2, -1e30f);
                tv[r] = t2;
                mx = fmaxf(mx, t2);
            }
            float ssum = 0.0f;
#pragma unroll
            for (int r = 0; r < KST; ++r) ssum += __expf(tv[r] - mx);
            na[j] = mx + __logf(ssum);
        }
#pragma unroll
        for (int j = 0; j < KST; ++j) alpha[j] = na[j];
    }

    // log_z = logsumexp(alpha + end)
    float mx = -3.0e38f;
#pragma unroll
    for (int j = 0; j < KST; ++j) mx = fmaxf(mx, alpha[j] + end_tr[j]);
    float ssum = 0.0f;
#pragma unroll
    for (int j = 0; j < KST; ++j) ssum += __expf(alpha[j] + end_tr[j] - mx);
    const float logz = mx + __logf(ssum);

    // score: t=0 term + chunk partials
    const int tag0 = tags[b * TLEN];
    float score = start_tr[tag0] + emissions[(size_t)b * TLEN * KST + tag0];
    if (TLEN >= 2 && mask[b * TLEN + 1] == 0) score += end_tr[tag0];  // length-1 edge
    for (int c = 0; c < NCHUNK; ++c) score += wsScore[c * BSZ + b];

    perBatch[b] = logz - score;
}

// ---------------------------------------------------------------------------
// Deterministic fixed-tree mean over 4096 per-batch values.
// ---------------------------------------------------------------------------
__global__ void __launch_bounds__(256) crf_reduce(const float* __restrict__ perBatch,
                                                  float* __restrict__ out)
{
    __shared__ float sm[256];
    const int tid = threadIdx.x;
    float s = 0.0f;
    for (int i = tid; i < BSZ; i += 256) s += perBatch[i];
    sm[tid] = s;
    __syncthreads();
    for (int off = 128; off > 0; off >>= 1) {
        if (tid < off) sm[tid] += sm[tid + off];
        __syncthreads();
    }
    if (tid == 0) out[0] = sm[0] * (1.0f / (float)BSZ);
}

extern "C" void kernel_launch(void* const* d_in, const int* in_sizes, int n_in,
                              void* d_out, int out_size, void* d_ws, size_t ws_size,
                              hipStream_t stream) {
    const float* emissions   = (const float*)d_in[0];
    const int*   mask        = (const int*)  d_in[1];
    const int*   tags        = (const int*)  d_in[2];
    const float* transitions = (const float*)d_in[3];
    const float* start_tr    = (const float*)d_in[4];
    const float* end_tr      = (const float*)d_in[5];

    float* ws       = (float*)d_ws;
    float* wsG      = ws;                                  // NCHUNK*B*64 floats (8 MB)
    float* wsScore  = wsG + (size_t)NCHUNK * BSZ * 64;     // NCHUNK*B floats
    float* perBatch = wsScore + (size_t)NCHUNK * BSZ;      // B floats

    // Phase 1: (B/4)*NCHUNK = 8192 waves, 8 waves/block -> 1024 blocks.
    crf_phase1<<<(BSZ / 4) * NCHUNK / 8, 256, 0, stream>>>(
        emissions, mask, tags, transitions, end_tr, wsG, wsScore);

    crf_phase2<<<(BSZ + 255) / 256, 256, 0, stream>>>(
        emissions, mask, tags, start_tr, end_tr, wsG, wsScore, perBatch);

    crf_reduce<<<1, 256, 0, stream>>>(perBatch, (float*)d_out);
}